// MCmodel_63067299774560
// MI455X (gfx1250) — compile-verified
//
#include <hip/hip_runtime.h>
#include <hip/hip_bf16.h>

// ---------------------------------------------------------------------------
// Problem constants (from reference)
// ---------------------------------------------------------------------------
#define NDATA   500000
#define NC      64
#define ND      32
#define KDIM    68          // 65 features (x^2[32], x[32], 1) padded to mult. of 4

// Radix sort config: 8-bit digits, 4 passes, stable LSD.
// CDNA5's 320 KB/WGP LDS lets us keep per-thread stable digit counters for
// 256 bins (256 threads x 256 bins x 4B = 256 KB).
#define NBINS       256
#define RDX_THREADS 256
#define EPT         16
#define CHUNK       (RDX_THREADS * EPT)                 // 4096
#define NCH         ((NDATA + CHUNK - 1) / CHUNK)       // 123 chunks per row

typedef float v8f __attribute__((ext_vector_type(8)));
typedef float v2f __attribute__((ext_vector_type(2)));

__device__ __forceinline__ unsigned flip_key(float f) {
    unsigned u = __float_as_uint(f);
    return u ^ ((u & 0x80000000u) ? 0xFFFFFFFFu : 0x80000000u);
}
__device__ __forceinline__ float unflip_key(unsigned u) {
    u ^= (u & 0x80000000u) ? 0x80000000u : 0xFFFFFFFFu;
    return __uint_as_float(u);
}

// ---------------------------------------------------------------------------
// Kernel 0: build B[KDIM][NC] and thresholds
//   B[d][c]      = w[c][d]                 (pairs with x^2)
//   B[32+d][c]   = -2 * w[c][d]*mu[c][d]   (pairs with x)
//   B[64][c]     = sum_d w*mu*mu           (pairs with 1)
//   B[65..67][c] = 0 (K padding)
// ---------------------------------------------------------------------------
__global__ void prep_kernel(const float* __restrict__ w,
                            const float* __restrict__ mu,
                            const float* __restrict__ profit,
                            float* __restrict__ Bmat,
                            float* __restrict__ thd) {
    int c = threadIdx.x;
    if (c < NC) {
        float cst = 0.0f;
        for (int d = 0; d < ND; ++d) {
            float wv = w[c * ND + d];
            float mv = mu[c * ND + d];
            Bmat[d * NC + c]        = wv;
            Bmat[(ND + d) * NC + c] = -2.0f * wv * mv;
            cst += wv * mv * mv;
        }
        Bmat[64 * NC + c] = cst;
        Bmat[65 * NC + c] = 0.0f;
        Bmat[66 * NC + c] = 0.0f;
        Bmat[67 * NC + c] = 0.0f;
        thd[c] = profit[c * 4 + 3];
    }
}

// ---------------------------------------------------------------------------
// Kernel 1: WMMA distance GEMM.  va = F(16xK) * B(Kx64), f32 16x16x4 WMMA.
// Block = 128 threads = 4 waves; each wave owns one 16-row data tile and all
// 64 cluster columns (4 accumulator tiles).  Results transposed through LDS
// and stored [C][N]-major as sortable keys + index payloads.
// ---------------------------------------------------------------------------
__global__ void __launch_bounds__(128)
dist_wmma_kernel(const float* __restrict__ data,
                 const float* __restrict__ Bmat,
                 unsigned* __restrict__ keys,
                 unsigned* __restrict__ pay) {
    __shared__ float sB[KDIM * NC];        // 17408 B, shared by all 4 waves
    __shared__ float sX[4 * 16 * 33];      // per-wave 16x32 data tile (pad 33)
    __shared__ float sT[4 * 64 * 17];      // per-wave 64x16 transposed result

    const int tid  = threadIdx.x;
    const int wave = tid >> 5;
    const int lane = tid & 31;
    const int half = lane >> 4;            // 0: lanes 0-15, 1: lanes 16-31
    const int l15  = lane & 15;

    // cooperative B load
    for (int i = tid; i < KDIM * NC; i += 128) sB[i] = Bmat[i];

    // tile of 16 data rows for this wave (N divides 16 exactly: 31250 tiles)
    int tile = blockIdx.x * 4 + wave;
    const int numTiles = NDATA / 16;       // 31250
    if (tile >= numTiles) tile = numTiles - 1;   // clamp: duplicate work, same values
    const int row0 = tile * 16;

    float* sXw = sX + wave * (16 * 33);
    for (int i = lane; i < 512; i += 32) {
        int r = i >> 5, d = i & 31;
        int grow = row0 + r;
        sXw[r * 33 + d] = (grow < NDATA) ? data[(size_t)grow * ND + d] : 0.0f;
    }
    __syncthreads();

    // accumulate: 17 K-steps x 4 column tiles of v_wmma_f32_16x16x4_f32
    v8f acc0 = {}, acc1 = {}, acc2 = {}, acc3 = {};
    #pragma unroll
    for (int k0 = 0; k0 < KDIM; k0 += 4) {
        const int ka = k0 + half * 2;      // A layout: v0 holds K=0/2, v1 holds K=1/3
        v2f a;
        // feature(row, k): k<32 -> x^2 ; k<64 -> x ; k==64 -> 1 ; else 0
        #pragma unroll
        for (int j = 0; j < 2; ++j) {
            const int k = ka + j;
            float f;
            if (k < 32)      { float x = sXw[l15 * 33 + k];      f = x * x; }
            else if (k < 64) { f = sXw[l15 * 33 + (k - 32)]; }
            else if (k == 64){ f = 1.0f; }
            else             { f = 0.0f; }
            a[j] = f;
        }
        v2f b0, b1, b2, b3;
        b0.x = sB[ka * NC +  0 + l15]; b0.y = sB[(ka + 1) * NC +  0 + l15];
        b1.x = sB[ka * NC + 16 + l15]; b1.y = sB[(ka + 1) * NC + 16 + l15];
        b2.x = sB[ka * NC + 32 + l15]; b2.y = sB[(ka + 1) * NC + 32 + l15];
        b3.x = sB[ka * NC + 48 + l15]; b3.y = sB[(ka + 1) * NC + 48 + l15];
        acc0 = __builtin_amdgcn_wmma_f32_16x16x4_f32(false, a, false, b0, (short)0, acc0, false, false);
        acc1 = __builtin_amdgcn_wmma_f32_16x16x4_f32(false, a, false, b1, (short)0, acc1, false, false);
        acc2 = __builtin_amdgcn_wmma_f32_16x16x4_f32(false, a, false, b2, (short)0, acc2, false, false);
        acc3 = __builtin_amdgcn_wmma_f32_16x16x4_f32(false, a, false, b3, (short)0, acc3, false, false);
    }

    // D layout: VGPR r, lanes0-15 -> (M=r, N=lane), lanes16-31 -> (M=8+r, N=lane-16)
    float* sTw = sT + wave * (64 * 17);
    const int m = 8 * half;
    #pragma unroll
    for (int r = 0; r < 8; ++r) {
        sTw[( 0 + l15) * 17 + (m + r)] = acc0[r];
        sTw[(16 + l15) * 17 + (m + r)] = acc1[r];
        sTw[(32 + l15) * 17 + (m + r)] = acc2[r];
        sTw[(48 + l15) * 17 + (m + r)] = acc3[r];
    }
    __syncthreads();

    // [C][N]-major store: per cluster, 16 consecutive n values are contiguous
    for (int i = lane; i < 1024; i += 32) {
        int c = i >> 4, r = i & 15;
        int grow = row0 + r;
        size_t o = (size_t)c * NDATA + grow;
        keys[o] = flip_key(sTw[c * 17 + r]);
        pay[o]  = (unsigned)grow;
    }
}

// ---------------------------------------------------------------------------
// Radix sort: histogram / scan / stable scatter (8-bit LSD, 4 passes)
// hist layout: hist[(c*NBINS + d) * NCH + chunk]
// ---------------------------------------------------------------------------
__global__ void __launch_bounds__(RDX_THREADS)
radix_hist_kernel(const unsigned* __restrict__ keys,
                  unsigned* __restrict__ hist, int shift) {
    const int c = blockIdx.y, chunk = blockIdx.x, t = threadIdx.x;
    __shared__ unsigned h[NBINS];
    h[t] = 0;                                   // RDX_THREADS == NBINS
    __syncthreads();
    const unsigned* row = keys + (size_t)c * NDATA;
    const int base = chunk * CHUNK;
    // prefetch the next chunk region (global_prefetch_b8 on gfx1250)
    if (base + CHUNK + t * 64 < NDATA)
        __builtin_prefetch(row + base + CHUNK + t * 16, 0, 1);
    for (int i = 0; i < EPT; ++i) {
        int n = base + t + i * RDX_THREADS;     // coalesced; order irrelevant for counts
        if (n < NDATA) atomicAdd(&h[(row[n] >> shift) & (NBINS - 1)], 1u);
    }
    __syncthreads();
    hist[((size_t)c * NBINS + t) * NCH + chunk] = h[t];
}

__global__ void __launch_bounds__(RDX_THREADS)
radix_scan_kernel(unsigned* __restrict__ hist) {
    // Exclusive scan of NBINS*NCH entries per row, bin-major order.
    // Two-level: per-bin serial scan over chunks (256 bins in parallel),
    // then scan of bin totals, then add-back.
    const int c = blockIdx.x;
    const int tot = NBINS * NCH;                // 31488
    __shared__ unsigned s[NBINS * NCH];         // 125952 B
    __shared__ unsigned btot[NBINS];
    unsigned* h = hist + (size_t)c * tot;
    for (int i = threadIdx.x; i < tot; i += RDX_THREADS) s[i] = h[i];
    __syncthreads();
    {   // per-bin exclusive scan over NCH chunks (one bin per thread)
        const int b = threadIdx.x;
        unsigned run = 0;
        for (int ch = 0; ch < NCH; ++ch) {
            unsigned v = s[b * NCH + ch];
            s[b * NCH + ch] = run;
            run += v;
        }
        btot[b] = run;
    }
    __syncthreads();
    if (threadIdx.x == 0) {                     // exclusive scan of 256 bin totals
        unsigned run = 0;
        for (int b = 0; b < NBINS; ++b) { unsigned v = btot[b]; btot[b] = run; run += v; }
    }
    __syncthreads();
    for (int i = threadIdx.x; i < tot; i += RDX_THREADS)
        h[i] = s[i] + btot[i / NCH];
    __syncthreads();
}

__global__ void __launch_bounds__(RDX_THREADS)
radix_scatter_kernel(const unsigned* __restrict__ skeys,
                     const unsigned* __restrict__ spay,
                     unsigned* __restrict__ dkeys,
                     unsigned* __restrict__ dpay,
                     const unsigned* __restrict__ hist, int shift) {
    const int c = blockIdx.y, chunk = blockIdx.x, t = threadIdx.x;
    // Per-thread stable digit counters: 256 threads x 256 bins = 256 KB LDS.
    // Fits only because CDNA5 WGPs have 320 KB of LDS.
    __shared__ unsigned cnts[NBINS * RDX_THREADS];   // 262144 B
    __shared__ unsigned binBase[NBINS];

    const unsigned* kr = skeys + (size_t)c * NDATA;
    const unsigned* pr = spay  + (size_t)c * NDATA;
    const int base = chunk * CHUNK + t * EPT;        // per-thread contiguous -> stability

    for (int d = 0; d < NBINS; ++d) cnts[d * RDX_THREADS + t] = 0;
    __syncthreads();

    unsigned k[EPT];
    #pragma unroll
    for (int i = 0; i < EPT; ++i) {
        int n = base + i;
        if (n < NDATA) {
            k[i] = kr[n];
            ++cnts[((k[i] >> shift) & (NBINS - 1)) * RDX_THREADS + t];
        } else k[i] = 0u;
    }
    __syncthreads();

    // exclusive prefix across threads within each digit (one digit per thread)
    {
        const int d = t;
        unsigned run = 0;
        for (int j = 0; j < RDX_THREADS; ++j) {
            unsigned v = cnts[d * RDX_THREADS + j];
            cnts[d * RDX_THREADS + j] = run;
            run += v;
        }
        binBase[d] = hist[((size_t)c * NBINS + d) * NCH + chunk];
    }
    __syncthreads();

    unsigned* dk = dkeys + (size_t)c * NDATA;
    unsigned* dp = dpay  + (size_t)c * NDATA;
    #pragma unroll
    for (int i = 0; i < EPT; ++i) {
        int n = base + i;
        if (n < NDATA) {
            unsigned key = k[i];
            int d = (key >> shift) & (NBINS - 1);
            unsigned pos = binBase[d] + cnts[d * RDX_THREADS + t]++;  // own column only
            dk[pos] = key;
            dp[pos] = pr[n];
        }
    }
}

// ---------------------------------------------------------------------------
// Finalize: unflip keys -> va_sorted, payload -> idx, mask = va <= thd[c]
// ---------------------------------------------------------------------------
__global__ void __launch_bounds__(256)
finalize_kernel(const unsigned* __restrict__ keys,
                const unsigned* __restrict__ pay,
                const float* __restrict__ thd,
                float* __restrict__ vout,
                int* __restrict__ iout,
                unsigned char* __restrict__ mout) {
    size_t i = (size_t)blockIdx.x * blockDim.x + threadIdx.x;
    const size_t total = (size_t)NC * NDATA;
    if (i < total) {
        float v = unflip_key(keys[i]);
        int c = (int)(i / NDATA);
        vout[i] = v;
        iout[i] = (int)pay[i];
        mout[i] = (v <= thd[c]) ? 1 : 0;
    }
}

// ---------------------------------------------------------------------------
// Launch
// ---------------------------------------------------------------------------
extern "C" void kernel_launch(void* const* d_in, const int* in_sizes, int n_in,
                              void* d_out, int out_size, void* d_ws, size_t ws_size,
                              hipStream_t stream) {
    const float* data   = (const float*)d_in[0];   // [N, 32]
    const float* mu     = (const float*)d_in[1];   // cluster [64, 32]
    const float* w      = (const float*)d_in[2];   // weight  [64, 32]
    const float* profit = (const float*)d_in[3];   // [64, 4]

    char* ws = (char*)d_ws;
    float*    Bmat = (float*)ws;                               // 68*64*4 = 17408 B
    float*    thd  = (float*)(ws + 17408);                     // 256 B
    unsigned* hist = (unsigned*)(ws + 17664);                  // 64*256*123*4 ~ 8 MB
    const size_t KP = (size_t)NC * NDATA * sizeof(unsigned);   // 128 MB each
    size_t off = (size_t)1 << 24;                              // 16 MB
    unsigned* keysA = (unsigned*)(ws + off); off += KP;
    unsigned* keysB = (unsigned*)(ws + off); off += KP;
    unsigned* payA  = (unsigned*)(ws + off); off += KP;
    unsigned* payB  = (unsigned*)(ws + off); off += KP;

    prep_kernel<<<1, 64, 0, stream>>>(w, mu, profit, Bmat, thd);

    const int numTiles = NDATA / 16;                 // 31250
    const int gemmBlocks = (numTiles + 3) / 4;       // 4 waves (tiles) per block
    dist_wmma_kernel<<<gemmBlocks, 128, 0, stream>>>(data, Bmat, keysA, payA);

    unsigned *sk = keysA, *sp = payA, *dk = keysB, *dp = payB;
    for (int pass = 0; pass < 4; ++pass) {
        const int shift = pass * 8;
        radix_hist_kernel<<<dim3(NCH, NC), RDX_THREADS, 0, stream>>>(sk, hist, shift);
        radix_scan_kernel<<<NC, RDX_THREADS, 0, stream>>>(hist);
        radix_scatter_kernel<<<dim3(NCH, NC), RDX_THREADS, 0, stream>>>(sk, sp, dk, dp, hist, shift);
        unsigned* tk = sk; sk = dk; dk = tk;
        unsigned* tp = sp; sp = dp; dp = tp;
    }
    // 4 passes -> final sorted data back in keysA/payA (== sk/sp)

    const size_t total = (size_t)NC * NDATA;
    float*         vout = (float*)d_out;
    int*           iout = (int*)(vout + total);
    unsigned char* mout = (unsigned char*)(iout + total);
    const int fblocks = (int)((total + 255) / 256);
    finalize_kernel<<<fblocks, 256, 0, stream>>>(sk, sp, thd, vout, iout, mout);
}